// _PerfectDenoiser_13400297963682
// MI455X (gfx1250) — compile-verified
//
#include <hip/hip_runtime.h>

// ---------------- problem constants (from the reference) ----------------
#define N_MAXC      128
#define NODE_VOCABC 128
#define EDGE_VOCABC 16
#define BATCHC      256
#define SEQC        8256
#define N_EDGESC    (SEQC - N_MAXC)                    // 8128

#define NODE_FLOATS ((size_t)BATCHC * N_MAXC * NODE_VOCABC)   // 4,194,304
#define EDGE_FLOATS ((size_t)BATCHC * N_EDGESC * EDGE_VOCABC) // 33,292,288
#define NODE_CHUNKS (unsigned)(NODE_FLOATS / 4)               // 1,048,576
#define EDGE_CHUNKS (unsigned)(EDGE_FLOATS / 4)               // 8,323,072
#define TOTAL_CHUNKS (NODE_CHUNKS + EDGE_CHUNKS)              // 9,371,648

// ---------------- CDNA5 async LDS->global store ----------------
#define __AS1 __attribute__((address_space(1)))
#define __AS3 __attribute__((address_space(3)))
typedef __attribute__((ext_vector_type(4))) int v4i;

#if defined(__has_builtin)
#  if __has_builtin(__builtin_amdgcn_global_store_async_from_lds_b128)
#    define HAVE_ASYNC_STORE 1
#  endif
#  if __has_builtin(__builtin_amdgcn_s_wait_asynccnt)
#    define HAVE_WAIT_ASYNC 1
#  endif
#endif

// One thread per 16-byte output chunk. Background chunks (-100 x4) stream
// from a 16-byte LDS constant via the gfx1250 async store path (ASYNCcnt);
// the single chunk holding the +100 is built in VGPRs and stored normally.
// The two paths never alias, so no ordering hazard exists between them.
__global__ __launch_bounds__(256) void _perfect_denoiser_fill_kernel(
    const int* __restrict__ x0, float* __restrict__ out) {
  __shared__ float bg[4];
  if (threadIdx.x < 4) bg[threadIdx.x] = -100.0f;
  __syncthreads();

  unsigned tid = blockIdx.x * 256u + threadIdx.x;
  if (tid >= TOTAL_CHUNKS) {
#if HAVE_WAIT_ASYNC
    __builtin_amdgcn_s_wait_asynccnt(0);
#endif
    return;
  }

  int v;        // one-hot index for this row
  int ci;       // which 4-float chunk of the row this thread owns
  size_t outOff;

  if (tid < NODE_CHUNKS) {
    // node region: row = b*128 + i, 32 chunks per 128-float row
    unsigned row = tid >> 5;          // b*N_MAX + i
    ci = (int)(tid & 31u);
    unsigned b = row >> 7;            // / N_MAX
    unsigned i = row & 127u;          // % N_MAX
    v = x0[(size_t)b * SEQC + i];
    outOff = (size_t)tid * 4;
  } else {
    // edge region: 4 chunks per 16-float row
    unsigned etid = tid - NODE_CHUNKS;
    unsigned b    = etid / (unsigned)(N_EDGESC * 4);   // const-div -> umulhi
    unsigned rem  = etid - b * (unsigned)(N_EDGESC * 4);
    unsigned e    = rem >> 2;
    ci = (int)(rem & 3u);
    v = x0[(size_t)b * SEQC + N_MAXC + e];
    outOff = NODE_FLOATS + (size_t)etid * 4;
  }

  float* dst = out + outOff;
  // Hot iff the one-hot element falls inside this 4-float chunk.
  // OOB / negative v matches no chunk -> all -100 (same as jax one_hot OOB).
  if ((v >> 2) == ci) {
    int s = v & 3;
    float4 val;
    val.x = (s == 0) ? 100.0f : -100.0f;
    val.y = (s == 1) ? 100.0f : -100.0f;
    val.z = (s == 2) ? 100.0f : -100.0f;
    val.w = (s == 3) ? 100.0f : -100.0f;
    *(float4*)dst = val;               // global_store_b128
  } else {
#if HAVE_ASYNC_STORE
    // gfx1250 async LDS->global: 16B per lane from the shared -100 pattern.
    __builtin_amdgcn_global_store_async_from_lds_b128(
        (__AS1 v4i*)dst, (__AS3 v4i*)bg, /*offset=*/0, /*cpol=*/0);
#else
    float4 val;
    val.x = -100.0f; val.y = -100.0f; val.z = -100.0f; val.w = -100.0f;
    *(float4*)dst = val;
#endif
  }

#if HAVE_WAIT_ASYNC
  __builtin_amdgcn_s_wait_asynccnt(0);
#endif
}

extern "C" void kernel_launch(void* const* d_in, const int* in_sizes, int n_in,
                              void* d_out, int out_size, void* d_ws, size_t ws_size,
                              hipStream_t stream) {
  // inputs: x0 (int), tokens (unused), pad_mask (unused), t (unused)
  const int* x0 = (const int*)d_in[0];
  float* out = (float*)d_out;

  const unsigned threads = 256;
  const unsigned blocks = (TOTAL_CHUNKS + threads - 1) / threads;  // 36,608
  _perfect_denoiser_fill_kernel<<<dim3(blocks), dim3(threads), 0, stream>>>(x0, out);
}